// economicgrasp_multi_88158498718150
// MI455X (gfx1250) — compile-verified
//
#include <hip/hip_runtime.h>
#include <hip/hip_bf16.h>

// ---------------------------------------------------------------------------
// economicgrasp point->voxel fuse for MI455X (gfx1250, wave32)
//
// Stages (all on `stream`, stream order gives dependencies):
//   1. init_ws        : zero sums/cnt, init hash table
//   2. build_hash     : key=(b<<24)|(x<<16)|(y<<8)|z per tgt row,
//                       hash insert with atomicMin(original index)
//                       == stable-argsort + searchsorted-left + order[pos]
//   3. scatter_points : wave-per-point, broadcast canonical voxel idx,
//                       2 coalesced atomicAdd f32 per lane (64 ch / 32 lanes)
//   4. fuse_gemm      : D(M,32) = [tgt_feats | sums/cnt](M,96) @ W(96,32) + b
//                       via V_WMMA_F32_16X16X4_F32, one 16x32 tile per wave;
//                       B fragments pre-packed in LDS in WMMA operand order
//                       so each step is a single ds_load_b64 per tile.
// ---------------------------------------------------------------------------

#define C_IMG   64
#define C_3D    32
#define CK      96              // C_3D + C_IMG
#define KSTEPS  (CK / 4)        // 24 WMMA K-steps
#define HBITS   19
#define HSIZE   (1u << HBITS)   // 524288 slots, load factor ~0.38 at M=200k
#define HMASK   (HSIZE - 1u)
#define EMPTY_KEY 0xFFFFFFFFu

typedef __attribute__((ext_vector_type(2))) float v2f;
typedef __attribute__((ext_vector_type(8))) float v8f;

__device__ __forceinline__ unsigned hash_of(unsigned k) {
    return (k * 2654435761u) >> (32u - HBITS);   // Fibonacci hash, top bits
}

// ---------------------------------------------------------------------------
__global__ void init_ws_kernel(float* __restrict__ sums, float* __restrict__ cnt,
                               unsigned* __restrict__ hkey, int* __restrict__ hidx,
                               int M) {
    const int stride = gridDim.x * blockDim.x;
    const int i0 = blockIdx.x * blockDim.x + threadIdx.x;
    const int nsum = M * C_IMG;
    for (int t = i0; t < nsum; t += stride) sums[t] = 0.0f;
    for (int t = i0; t < M; t += stride) cnt[t] = 0.0f;
    for (unsigned t = (unsigned)i0; t < HSIZE; t += (unsigned)stride) {
        hkey[t] = EMPTY_KEY;
        hidx[t] = 0x7FFFFFFF;
    }
}

// ---------------------------------------------------------------------------
__global__ void build_hash_kernel(const int* __restrict__ tgt_coords,
                                  unsigned* __restrict__ hkey,
                                  int* __restrict__ hidx, int M) {
    int j = blockIdx.x * blockDim.x + threadIdx.x;
    if (j >= M) return;
    const int* tc = tgt_coords + 4 * (size_t)j;
    unsigned key = ((unsigned)tc[0] << 24) | ((unsigned)tc[1] << 16) |
                   ((unsigned)tc[2] << 8)  |  (unsigned)tc[3];
    unsigned slot = hash_of(key);
    while (true) {
        unsigned prev = atomicCAS(&hkey[slot], EMPTY_KEY, key);
        if (prev == EMPTY_KEY || prev == key) {
            atomicMin(&hidx[slot], j);   // min original index among equal keys
            break;
        }
        slot = (slot + 1u) & HMASK;
    }
}

__device__ __forceinline__ int hash_lookup(const unsigned* __restrict__ hkey,
                                           const int* __restrict__ hidx,
                                           unsigned key) {
    unsigned slot = hash_of(key);
    for (unsigned probe = 0; probe < HSIZE; ++probe) {
        unsigned k = hkey[slot];
        if (k == key) return hidx[slot];
        if (k == EMPTY_KEY) return 0;    // cannot happen: every point maps to a tgt row
        slot = (slot + 1u) & HMASK;
    }
    return 0;
}

// ---------------------------------------------------------------------------
// One wave32 per point: lanes cooperatively scatter the 64-channel feature.
__global__ void scatter_points_kernel(const float* __restrict__ pfeat,
                                      const int* __restrict__ pts_idx,
                                      const int* __restrict__ tgt_coords,
                                      const unsigned* __restrict__ hkey,
                                      const int* __restrict__ hidx,
                                      float* __restrict__ sums,
                                      float* __restrict__ cnt, int BN) {
    int gid  = blockIdx.x * blockDim.x + threadIdx.x;
    int wave = gid >> 5;
    int lane = gid & 31;
    if (wave >= BN) return;

    int j = pts_idx[wave];                       // uniform across the wave
    const int* tc = tgt_coords + 4 * (size_t)j;  // broadcast load
    unsigned key = ((unsigned)tc[0] << 24) | ((unsigned)tc[1] << 16) |
                   ((unsigned)tc[2] << 8)  |  (unsigned)tc[3];
    // canonical voxel index; force scalar so sums base addressing is SALU
    int c = __builtin_amdgcn_readfirstlane(hash_lookup(hkey, hidx, key));

    const float* pf = pfeat + (size_t)wave * C_IMG;
    float* dst = sums + (size_t)c * C_IMG;
    atomicAdd(&dst[lane],      pf[lane]);        // coalesced 128B per half
    atomicAdd(&dst[lane + 32], pf[lane + 32]);
    if (lane == 0) atomicAdd(&cnt[c], 1.0f);
}

// ---------------------------------------------------------------------------
// Fuse GEMM: each wave computes a 16x32 output tile with f32 WMMA.
//   A tile (16x4 f32):  lane L -> row = L%16, K-pair base = (L/16)*2
//   B tile (4x16 f32):  lane L -> col = L%16, K rows {base, base+1}
//   C/D (16x16 f32):    VGPR v -> row = v + (L<16 ? 0 : 8), col = L%16
// B fragments are pre-packed in LDS as v2f in exactly the per-lane operand
// order, so the inner loop issues one ds_load_b64 per fragment (no shuffles).
__global__ void fuse_gemm_kernel(const float* __restrict__ tgt_feats,
                                 const float* __restrict__ W,      // (96,32) row-major
                                 const float* __restrict__ bias,   // (32,)
                                 const float* __restrict__ sums,
                                 const float* __restrict__ cnt,
                                 float* __restrict__ out, int M) {
    __shared__ v2f  sWp[KSTEPS * 2 * 32];   // [step][tile][lane] -> 12 KB
    __shared__ float sB[C_3D];

    for (int i = threadIdx.x; i < KSTEPS * 2 * 32; i += blockDim.x) {
        const int step  = i >> 6;            // / 64
        const int tile  = (i >> 5) & 1;
        const int plane = i & 31;
        const int pcol  = (plane & 15) + 16 * tile;
        const int pk0   = step * 4 + ((plane >> 4) << 1);
        v2f val;
        val.x = W[(size_t)pk0 * C_3D + pcol];
        val.y = W[(size_t)(pk0 + 1) * C_3D + pcol];
        sWp[i] = val;
    }
    for (int i = threadIdx.x; i < C_3D; i += blockDim.x) sB[i] = bias[i];
    __syncthreads();

    const int lane = threadIdx.x & 31;
    const int wave = (blockIdx.x * blockDim.x + threadIdx.x) >> 5;
    const int m0 = wave * 16;
    if (m0 >= M) return;

    const int rowA = lane & 15;          // this lane's A row within the tile
    const int koff = (lane >> 4) << 1;   // 0 for lanes 0-15, 2 for lanes 16-31
    const int col  = lane & 15;          // this lane's B/D column within a tile
    const int m    = m0 + rowA;

    float rc = cnt[m];
    rc = 1.0f / fmaxf(rc, 1.0f);         // scatter-mean normalizer (exact, as ref)
    const float* tf = tgt_feats + (size_t)m * C_3D;
    const float* sm = sums      + (size_t)m * C_IMG;

    v8f acc0 = {};                        // columns 0..15
    v8f acc1 = {};                        // columns 16..31

    #pragma unroll
    for (int ks = 0; ks < CK; ks += 4) {  // 24 K-steps of 4
        const int k0 = ks + koff;         // branch below uniform after unroll
        v2f a;
        a.x = (k0     < C_3D) ? tf[k0]     : sm[k0     - C_3D] * rc;
        a.y = (k0 + 1 < C_3D) ? tf[k0 + 1] : sm[k0 + 1 - C_3D] * rc;

        const v2f b0 = sWp[(ks >> 2) * 64 + lane];        // tile 0, ds_load_b64
        const v2f b1 = sWp[(ks >> 2) * 64 + 32 + lane];   // tile 1, ds_load_b64

        acc0 = __builtin_amdgcn_wmma_f32_16x16x4_f32(
            false, a, false, b0, (short)0, acc0, false, false);
        acc1 = __builtin_amdgcn_wmma_f32_16x16x4_f32(
            false, a, false, b1, (short)0, acc1, false, false);
    }

    const int rowBase = m0 + ((lane >> 4) << 3);  // +8 for hi half-wave
    const float bias0 = sB[col];
    const float bias1 = sB[col + 16];
    #pragma unroll
    for (int v = 0; v < 8; ++v) {
        float* orow = out + (size_t)(rowBase + v) * C_3D;
        orow[col]      = acc0[v] + bias0;
        orow[col + 16] = acc1[v] + bias1;
    }
}

// ---------------------------------------------------------------------------
extern "C" void kernel_launch(void* const* d_in, const int* in_sizes, int n_in,
                              void* d_out, int out_size, void* d_ws, size_t ws_size,
                              hipStream_t stream) {
    const float* pfeat      = (const float*)d_in[0];   // (BN, 64)
    const float* tgt_feats  = (const float*)d_in[1];   // (M, 32)
    const float* W_fuse     = (const float*)d_in[2];   // (96, 32)
    const float* b_fuse     = (const float*)d_in[3];   // (32,)
    const int*   tgt_coords = (const int*)d_in[4];     // (M, 4)
    const int*   pts_idx    = (const int*)d_in[5];     // (BN,)
    float* out = (float*)d_out;

    const int M  = in_sizes[4] / 4;
    const int BN = in_sizes[5];

    // workspace layout: sums | cnt | hkey | hidx
    char* ws = (char*)d_ws;
    float*    sums = (float*)ws;                 ws += (size_t)M * C_IMG * sizeof(float);
    float*    cnt  = (float*)ws;                 ws += (size_t)M * sizeof(float);
    unsigned* hkey = (unsigned*)ws;              ws += (size_t)HSIZE * sizeof(unsigned);
    int*      hidx = (int*)ws;

    init_ws_kernel<<<2048, 256, 0, stream>>>(sums, cnt, hkey, hidx, M);
    build_hash_kernel<<<(M + 255) / 256, 256, 0, stream>>>(tgt_coords, hkey, hidx, M);

    const long long scatter_threads = (long long)BN * 32;
    scatter_points_kernel<<<(int)((scatter_threads + 255) / 256), 256, 0, stream>>>(
        pfeat, pts_idx, tgt_coords, hkey, hidx, sums, cnt, BN);

    const long long gemm_threads = (long long)((M + 15) / 16) * 32;
    fuse_gemm_kernel<<<(int)((gemm_threads + 255) / 256), 256, 0, stream>>>(
        tgt_feats, W_fuse, b_fuse, sums, cnt, out, M);
}